// StyleLoss_17961553232007
// MI455X (gfx1250) — compile-verified
//
#include <hip/hip_runtime.h>
#include <hip/hip_bf16.h>

typedef float v2f __attribute__((ext_vector_type(2)));
typedef float v8f __attribute__((ext_vector_type(8)));

static constexpr int Bdim = 256;   // rows
static constexpr int Ddim = 512;   // cols
static constexpr int TILES = Bdim / 16;  // 16 row-tiles of 16 rows each
static constexpr int KSPLIT = 8;         // preferred k-parallelism

// Select element i (runtime) from a v8f without scratch: cndmask chain.
__device__ __forceinline__ float sel8(v8f v, int i) {
    float r = v[0];
    r = (i == 1) ? v[1] : r;
    r = (i == 2) ? v[2] : r;
    r = (i == 3) ? v[3] : r;
    r = (i == 4) ? v[4] : r;
    r = (i == 5) ? v[5] : r;
    r = (i == 6) ? v[6] : r;
    r = (i == 7) ? v[7] : r;
    return r;
}

// One wave (32 lanes) per block. Block index encodes (ksplit, tile).
// Computes partial ax = ||x_r||^2, ay = ||y_r||^2, c = <x_r,y_r> over a
// compile-time k-subrange for a 16-row tile, via the diagonals of
// XX^T / YY^T / XY^T accumulated with V_WMMA_F32_16X16X4_F32.
// KCHUNK is a template parameter so the loop fully unrolls and the
// global_load_b64s can all be issued ahead of the WMMA chain (MLP).
template <int KCHUNK>
__global__ __launch_bounds__(32) void gram_partial_kernel(
    const float* __restrict__ X, const float* __restrict__ Y,
    float* __restrict__ ws)
{
    const int tile  = blockIdx.x % TILES;
    const int split = blockIdx.x / TILES;

    const int lane = threadIdx.x;          // 0..31, full wave active (EXEC all 1s)
    const int m16  = lane & 15;            // row within tile
    const int koff = (lane >> 4) << 1;     // lanes 0-15 -> K=0,1 ; lanes 16-31 -> K=2,3
    const int row  = tile * 16 + m16;

    const float* px = X + (size_t)row * Ddim + koff + split * KCHUNK;
    const float* py = Y + (size_t)row * Ddim + koff + split * KCHUNK;

    constexpr int NIT = KCHUNK / 4;        // wmma k-steps in this chunk

    // Stage all loads first: maximize loads in flight before consuming.
    v2f a[NIT], b[NIT];
#pragma unroll
    for (int i = 0; i < NIT; ++i) {
        a[i] = *(const v2f*)(px + 4 * i);
        b[i] = *(const v2f*)(py + 4 * i);
    }

    v8f axx = {};   // accumulates X * X^T  (16x16 tile)
    v8f ayy = {};   // accumulates Y * Y^T
    v8f axy = {};   // accumulates X * Y^T

#pragma unroll
    for (int i = 0; i < NIT; ++i) {
        // Row-striped B layout for the transposed tile coincides with the A
        // layout registers, so A==B operand gives the symmetric product.
        axx = __builtin_amdgcn_wmma_f32_16x16x4_f32(
                  false, a[i], false, a[i], (short)0, axx, false, false);
        ayy = __builtin_amdgcn_wmma_f32_16x16x4_f32(
                  false, b[i], false, b[i], (short)0, ayy, false, false);
        axy = __builtin_amdgcn_wmma_f32_16x16x4_f32(
                  false, a[i], false, b[i], (short)0, axy, false, false);
    }

    // Diagonal of the 16x16 f32 C/D layout:
    //   lanes 0..7  : VGPR v = lane   -> (M,N) = (lane, lane)
    //   lanes 24..31: VGPR v = lane-24 -> (M,N) = (lane-16, lane-16)
    const bool valid = (lane < 8) || (lane >= 24);
    const int  idx   = (lane < 8) ? lane : (lane - 24);
    const int  m     = (lane < 8) ? lane : (lane - 16);

    const float ax = sel8(axx, idx);
    const float ay = sel8(ayy, idx);
    const float c  = sel8(axy, idx);

    if (valid) {
        float* w = ws + (((size_t)split * TILES + tile) * 16 + m) * 3;
        w[0] = ax;
        w[1] = ay;
        w[2] = c;
    }
}

// Single block: sum k-splits per row in fixed order, combine per-row terms,
// fixed-order LDS tree reduction -> bitwise deterministic scalar output.
__global__ __launch_bounds__(256) void gram_finish_kernel(
    const float* __restrict__ ws, float* __restrict__ out,
    int ksplit, float scale)
{
    __shared__ float red[Bdim];
    const int t    = threadIdx.x;      // 0..255, one per row
    const int tile = t >> 4;
    const int m    = t & 15;

    float ax = 0.f, ay = 0.f, c = 0.f;
    for (int s = 0; s < ksplit; ++s) {
        const float* w = ws + (((size_t)s * TILES + tile) * 16 + m) * 3;
        ax += w[0];
        ay += w[1];
        c  += w[2];
    }
    red[t] = ax * ax + ay * ay - 2.f * c * c;
    __syncthreads();

#pragma unroll
    for (int off = 128; off > 0; off >>= 1) {
        if (t < off) red[t] += red[t + off];
        __syncthreads();
    }
    if (t == 0) out[0] = red[0] * scale;
}

extern "C" void kernel_launch(void* const* d_in, const int* in_sizes, int n_in,
                              void* d_out, int out_size, void* d_ws, size_t ws_size,
                              hipStream_t stream) {
    (void)in_sizes; (void)n_in; (void)out_size;
    const float* X = (const float*)d_in[0];
    const float* Y = (const float*)d_in[1];
    float* out = (float*)d_out;
    float* ws  = (float*)d_ws;

    // loss = scale * sum_r (||x_r||^4 + ||y_r||^4 - 2 <x_r,y_r>^2)
    // scale = 1/(b*d*d) * (1/(2 b d))^2
    const double bd = (double)Bdim * (double)Ddim;
    const float scale =
        (float)(1.0 / ((double)Bdim * Ddim * Ddim) / (4.0 * bd * bd));

    // Deterministic ksplit choice based on available workspace.
    if (ws_size >= (size_t)KSPLIT * TILES * 16 * 3 * sizeof(float)) {
        gram_partial_kernel<Ddim / KSPLIT>
            <<<KSPLIT * TILES, 32, 0, stream>>>(X, Y, ws);
        gram_finish_kernel<<<1, 256, 0, stream>>>(ws, out, KSPLIT, scale);
    } else {
        gram_partial_kernel<Ddim>
            <<<TILES, 32, 0, stream>>>(X, Y, ws);
        gram_finish_kernel<<<1, 256, 0, stream>>>(ws, out, 1, scale);
    }
}